// CustomModelEmbeddingBagGroup_13993003451117
// MI455X (gfx1250) — compile-verified
//
#include <hip/hip_runtime.h>

// ---------------------------------------------------------------------------
// CustomModelEmbeddingBagGroup on MI455X (gfx1250, wave32)
//
// Math:  out = sum_i sum_d (5*W0[idx_i,d] + 10*W1[idx_i,d] + 6*W2[idx_i,d])
// (offsets cancel: summing segment sums over all bags == summing all rows)
//
// Strategy: histogram indices (V counters in L2), then stream each table row
// exactly once; row-sums over D=128 done with V_WMMA_F32_16X16X4_F32 using a
// ones A-matrix, accumulated in f32.  ~157 MB total traffic vs 1.26 GB naive.
// ---------------------------------------------------------------------------

typedef __attribute__((ext_vector_type(2))) float v2f;
typedef __attribute__((ext_vector_type(8))) float v8f;

#define D_DIM 128

// Kernel 1: zero the histogram counters and the scalar output.
__global__ void ebg_zero_kernel(unsigned int* __restrict__ cnt,
                                float* __restrict__ out, int v) {
  int i = blockIdx.x * blockDim.x + threadIdx.x;
  if (i < v) cnt[i] = 0u;
  if (i == 0) out[0] = 0.0f;
}

// Kernel 2: histogram of indices. Avg ~8 hits per bucket over 100k buckets ->
// very low contention; u32 atomics resolve in L2.
__global__ void ebg_hist_kernel(const int* __restrict__ idx,
                                unsigned int* __restrict__ cnt, int n) {
  int i = blockIdx.x * blockDim.x + threadIdx.x;
  if (i < n) {
    atomicAdd(&cnt[idx[i]], 1u);
  }
}

// Kernel 3: each wave owns 16 table rows. Combine the three tables with the
// group weights elementwise, then row-sum over D=128 via 32 chained
// v_wmma_f32_16x16x4_f32 ops (A = ones). Scale by histogram count, reduce.
__global__ void ebg_rowsum_wmma_kernel(const float* __restrict__ W0,
                                       const float* __restrict__ W1,
                                       const float* __restrict__ W2,
                                       const unsigned int* __restrict__ cnt,
                                       float* __restrict__ out, int v) {
  const int lane    = threadIdx.x & 31;
  const int waveId  = (blockIdx.x * blockDim.x + threadIdx.x) >> 5;
  const int row0    = waveId * 16;

  // B-matrix (4x16, K x N) lane mapping for 32-bit data:
  //   lanes 0-15 : K = k0+0 (vgpr0), k0+1 (vgpr1), N = lane
  //   lanes 16-31: K = k0+2 (vgpr0), k0+3 (vgpr1), N = lane-16
  const int rlocal  = lane & 15;          // which table row of this tile
  const int colBase = (lane >> 4) << 1;   // 0 or 2
  int row  = row0 + rlocal;
  int rowc = (row < v) ? row : (v - 1);   // clamp: keeps EXEC all-ones

  const float2* __restrict__ p0 =
      reinterpret_cast<const float2*>(W0 + (size_t)rowc * D_DIM + colBase);
  const float2* __restrict__ p1 =
      reinterpret_cast<const float2*>(W1 + (size_t)rowc * D_DIM + colBase);
  const float2* __restrict__ p2 =
      reinterpret_cast<const float2*>(W2 + (size_t)rowc * D_DIM + colBase);

  v2f ones;
  ones[0] = 1.0f;
  ones[1] = 1.0f;
  v8f acc = {};

#pragma unroll
  for (int k0 = 0; k0 < D_DIM; k0 += 4) {
    const int e = k0 >> 1;                // float2 element index (byte ofs k0*4)
    float2 a = p0[e];
    float2 b = p1[e];
    float2 c = p2[e];
    v2f comb;
    comb[0] = fmaf(5.0f, a.x, fmaf(10.0f, b.x, 6.0f * c.x));
    comb[1] = fmaf(5.0f, a.y, fmaf(10.0f, b.y, 6.0f * c.y));
    // D[m][n] = sum_k 1 * comb[k][n] + C  -> accumulates column sums of the
    // 4x16 tile == per-table-row partial sums, identical across all m.
    acc = __builtin_amdgcn_wmma_f32_16x16x4_f32(
        /*neg_a=*/false, ones, /*neg_b=*/false, comb,
        /*c_mod=*/(short)0, acc, /*reuse_a=*/false, /*reuse_b=*/false);
  }

  // C/D layout: vgpr0 holds M=0 (lanes 0-15, N=lane). All M rows identical, so
  // take lanes 0-15 only; lanes 16-31 hold a duplicate (M=8) -> contribute 0.
  int   myRow = row0 + lane;
  float c     = (lane < 16 && myRow < v) ? (float)cnt[myRow] : 0.0f;
  float val   = acc[0] * c;

  // wave32 tree reduction
#pragma unroll
  for (int off = 16; off > 0; off >>= 1) {
    val += __shfl_down(val, off, 32);
  }
  if (lane == 0) {
    atomicAdd(out, val);   // global_atomic_add_f32; ~6250 waves total
  }
}

extern "C" void kernel_launch(void* const* d_in, const int* in_sizes, int n_in,
                              void* d_out, int out_size, void* d_ws, size_t ws_size,
                              hipStream_t stream) {
  const int*   eb_input = (const int*)d_in[0];
  // d_in[1] = eb_offset: mathematically irrelevant (bags are summed away)
  const float* W0 = (const float*)d_in[2];
  const float* W1 = (const float*)d_in[3];
  const float* W2 = (const float*)d_in[4];
  float*       out = (float*)d_out;
  unsigned int* cnt = (unsigned int*)d_ws;   // V * 4 bytes = 400 KB scratch

  const int N = in_sizes[0];
  const int V = in_sizes[2] / D_DIM;

  ebg_zero_kernel<<<(V + 255) / 256, 256, 0, stream>>>(cnt, out, V);
  ebg_hist_kernel<<<(N + 255) / 256, 256, 0, stream>>>(eb_input, cnt, N);

  const int waves  = (V + 15) / 16;          // one wave per 16-row tile
  const int blocks = (waves + 7) / 8;        // 256 threads = 8 waves / block
  ebg_rowsum_wmma_kernel<<<blocks, 256, 0, stream>>>(W0, W1, W2, cnt, out, V);
}